// LogicConv2d_74457553044329
// MI455X (gfx1250) — compile-verified
//
#include <hip/hip_runtime.h>

// ---- problem constants (from reference) ----
#define BB   16
#define CIN  32
#define HH   64
#define WW   64
#define KK   64
#define NN   8
#define PAD  2
// tile of output pixels handled per block
#define TH   16
#define TW   16
#define SH   (TH + 4)          // 20 slab rows   (RF=5 -> +4)
#define SW   (TW + 4)          // 20 slab cols
#define SLAB (CIN * SH * SW)   // 12800 floats = 51.2 KB LDS

typedef float v2f __attribute__((ext_vector_type(2)));
typedef float v8f __attribute__((ext_vector_type(8)));

// _C gate table (16 gates x 4 coefficients)
__device__ __constant__ float cC[16][4] = {
    {0,0,0,0},{0,0,0,1},{0,1,0,-1},{0,1,0,0},
    {0,0,1,-1},{0,0,1,0},{0,1,1,-2},{0,1,1,-1},
    {1,-1,-1,1},{1,-1,-1,2},{1,0,-1,0},{1,0,-1,1},
    {1,-1,0,0},{1,-1,0,1},{1,0,0,-1},{1,0,0,0}};

// -------------------------------------------------------------------------
// Kernel 1: coefficient precompute via V_WMMA_F32_16X16X4_F32 (exact f32).
// 960 tree nodes total: level bases {0,512,768,896}, row = base + n*K + k.
// Each wave handles 16 nodes (one WMMA M-tile); contraction over 16 gates
// is 4 accumulating K=4 WMMAs. coef[row*4 + c] = sum_g softmax(logits)[g]*C[g][c]
// -------------------------------------------------------------------------
__global__ __launch_bounds__(32) void coef_kernel(
    const float* __restrict__ w0, const float* __restrict__ w1,
    const float* __restrict__ w2, const float* __restrict__ w3,
    float* __restrict__ coef)
{
  const int tile = blockIdx.x;        // 0..59
  const int lane = threadIdx.x;       // 0..31
  const int m    = lane & 15;         // matrix row within tile
  const int hi   = lane >> 4;         // half-wave selector
  const int row  = tile * 16 + m;     // global node row 0..959

  const float* lg;
  if (row < 512)      lg = w0 + row * 16;
  else if (row < 768) lg = w1 + (row - 512) * 16;
  else if (row < 896) lg = w2 + (row - 768) * 16;
  else                lg = w3 + (row - 896) * 16;

  // per-lane softmax of this node's 16 logits (TEMPERATURE == 1)
  float s[16];
  float mx = lg[0];
#pragma unroll
  for (int g = 0; g < 16; ++g) { s[g] = lg[g]; mx = fmaxf(mx, s[g]); }
  float sum = 0.f;
#pragma unroll
  for (int g = 0; g < 16; ++g) { s[g] = __expf(s[g] - mx); sum += s[g]; }
  const float inv = 1.f / sum;
#pragma unroll
  for (int g = 0; g < 16; ++g) s[g] *= inv;

  const int col = m;                  // B/D column index = lane & 15
  v8f d = {};                         // f32 accumulator (16x16)

#pragma unroll
  for (int j = 0; j < 4; ++j) {       // gate chunks of 4 (K dimension)
    v2f a, b;
    // A 16x4 layout: lanes 0-15 -> K={0,1}, lanes 16-31 -> K={2,3}
    a.x = hi ? s[4 * j + 2] : s[4 * j + 0];
    a.y = hi ? s[4 * j + 3] : s[4 * j + 1];
    // B 4x16 layout: lanes 0-15 -> rows {0,1}, lanes 16-31 -> rows {2,3}
    const int kk0 = 4 * j + 2 * hi;
    b.x = (col < 4) ? cC[kk0 + 0][col] : 0.f;
    b.y = (col < 4) ? cC[kk0 + 1][col] : 0.f;
    d = __builtin_amdgcn_wmma_f32_16x16x4_f32(
            false, a, false, b, (short)0, d, false, false);
  }

  // D 16x16 layout: vgpr v -> row v (lanes 0-15) / row v+8 (lanes 16-31)
#pragma unroll
  for (int v = 0; v < 8; ++v) {
    if (col < 4) {
      const int node = tile * 16 + v + hi * 8;
      coef[node * 4 + col] = d[v];
    }
  }
}

// -------------------------------------------------------------------------
// Kernel 2: fold idx_c/idx_h/idx_w into per-(k,leaf) LDS slab offsets.
// leaf = p*8 + n, p in {0,1} (a/b operand), n in 0..7
// -------------------------------------------------------------------------
__global__ void off_kernel(const int* __restrict__ idx_h,
                           const int* __restrict__ idx_w,
                           const int* __restrict__ idx_c,
                           int* __restrict__ off)
{
  const int t = blockIdx.x * blockDim.x + threadIdx.x;
  if (t >= KK * 16) return;
  const int k = t >> 4, leaf = t & 15, p = leaf >> 3, n = leaf & 7;
  const int i = (p * KK + k) * NN + n;
  off[t] = idx_c[i] * (SH * SW) + idx_h[i] * SW + idx_w[i];
}

// -------------------------------------------------------------------------
// Kernel 3: main gather + gate-tree evaluation.
// Block = one (batch, 16x16 pixel tile); stage 32x20x20 input slab in LDS,
// loop over all 64 k (uniform -> coef/offset reads lower to scalar loads).
// -------------------------------------------------------------------------
__global__ __launch_bounds__(256) void logic_conv_kernel(
    const float* __restrict__ x,
    const float* __restrict__ coef,   // 960*4 floats
    const int*   __restrict__ off,    // 64*16 ints
    float* __restrict__ out)
{
  __shared__ float slab[SLAB];

  const int b   = blockIdx.z;
  const int oh0 = blockIdx.y * TH;    // padded-space slab origin == output row
  const int ow0 = blockIdx.x * TW;
  const int tid = threadIdx.x;

  // stage slab: padded rows [oh0, oh0+SH), padded cols [ow0, ow0+SW)
  const float* xb = x + b * (CIN * HH * WW);
  for (int i = tid; i < SLAB; i += 256) {
    const int c   = i / (SH * SW);
    const int rem = i - c * (SH * SW);
    const int r   = rem / SW;
    const int w   = rem - r * SW;
    const int gr  = oh0 + r - PAD;    // unpadded coords
    const int gc  = ow0 + w - PAD;
    float v = 0.f;
    if ((unsigned)gr < (unsigned)HH && (unsigned)gc < (unsigned)WW)
      v = xb[(c * HH + gr) * WW + gc];
    slab[i] = v;
  }
  __syncthreads();

  const int ohl  = tid >> 4;          // 0..15 local pixel row
  const int owl  = tid & 15;          // 0..15 local pixel col
  const int base = ohl * SW + owl;

  float* outp = out + ((b * KK) * HH + (oh0 + ohl)) * WW + (ow0 + owl);

  for (int k = 0; k < KK; ++k) {
    const int* offk = off + k * 16;

    float v[16];
#pragma unroll
    for (int l = 0; l < 16; ++l) v[l] = slab[offk[l] + base];

    // level 0: 8 gates, rows n*64+k
    float cur[8];
#pragma unroll
    for (int n = 0; n < 8; ++n) {
      const float* c4 = coef + (n * KK + k) * 4;
      const float a = v[n], bq = v[8 + n];
      cur[n] = fmaf(c4[3], a * bq, fmaf(c4[2], bq, fmaf(c4[1], a, c4[0])));
    }
    // level 1: 4 gates, rows 512 + n*64 + k
    float n1[4];
#pragma unroll
    for (int n = 0; n < 4; ++n) {
      const float* c4 = coef + (512 + n * KK + k) * 4;
      const float a = cur[2 * n], bq = cur[2 * n + 1];
      n1[n] = fmaf(c4[3], a * bq, fmaf(c4[2], bq, fmaf(c4[1], a, c4[0])));
    }
    // level 2: 2 gates, rows 768 + n*64 + k
    float n2[2];
#pragma unroll
    for (int n = 0; n < 2; ++n) {
      const float* c4 = coef + (768 + n * KK + k) * 4;
      const float a = n1[2 * n], bq = n1[2 * n + 1];
      n2[n] = fmaf(c4[3], a * bq, fmaf(c4[2], bq, fmaf(c4[1], a, c4[0])));
    }
    // level 3: 1 gate, row 896 + k
    {
      const float* c4 = coef + (896 + k) * 4;
      const float a = n2[0], bq = n2[1];
      const float r = fmaf(c4[3], a * bq, fmaf(c4[2], bq, fmaf(c4[1], a, c4[0])));
      outp[k * (HH * WW)] = r;
    }
  }
}

// -------------------------------------------------------------------------
extern "C" void kernel_launch(void* const* d_in, const int* in_sizes, int n_in,
                              void* d_out, int out_size, void* d_ws, size_t ws_size,
                              hipStream_t stream)
{
  (void)in_sizes; (void)n_in; (void)out_size; (void)ws_size;
  const float* x     = (const float*)d_in[0];
  const int*   idx_h = (const int*)d_in[1];
  const int*   idx_w = (const int*)d_in[2];
  const int*   idx_c = (const int*)d_in[3];
  const float* w0    = (const float*)d_in[4];
  const float* w1    = (const float*)d_in[5];
  const float* w2    = (const float*)d_in[6];
  const float* w3    = (const float*)d_in[7];
  float* out = (float*)d_out;

  float* coef = (float*)d_ws;                    // 960*4 floats (15360 B)
  int*   off  = (int*)((char*)d_ws + 16384);     // 1024 ints

  coef_kernel<<<60, 32, 0, stream>>>(w0, w1, w2, w3, coef);
  off_kernel<<<4, 256, 0, stream>>>(idx_h, idx_w, idx_c, off);
  logic_conv_kernel<<<dim3(WW / TW, HH / TH, BB), 256, 0, stream>>>(x, coef, off, out);
}